// ModuleCorrelation_34651796144461
// MI455X (gfx1250) — compile-verified
//
#include <hip/hip_runtime.h>

typedef float v2f __attribute__((ext_vector_type(2)));
typedef float v8f __attribute__((ext_vector_type(8)));

#define BATCH  4
#define CH     256
#define HDIM   128
#define WDIM   128
#define MAXD   4
#define NDISP  9           // displacements per axis
#define NOUT   81          // 9*9 displacement planes
#define REG    8           // 8x8 pixel region per workgroup (4 waves)
#define NPIXR  64          // pixels per region
#define NPIX   16          // pixels per wave tile (4x4)
#define QW     12          // per-wave patch width  = 4 + 2*MAXD
#define QWR    16          // shared patch width    = 8 + 2*MAXD
#define NQR    256         // shared patch positions (16x16)
#define KC     32          // channels staged per chunk
#define NCHUNK (CH / KC)
#define AS     34          // padded LDS row stride (even -> aligned b64)
#define BS     34
#define DSTR   145

// Per wave: D[p,q] = sum_c x1tile[p,c] * x2patch[q,c] as a 16x144x256 GEMM via
// 9 accumulator tiles of V_WMMA_F32_16X16X4_F32 (exact fp32).
// 4 waves per WG share one 16x16 x2 patch (2.25x less staging traffic).
__global__ __launch_bounds__(128) void ModuleCorrelation_34651796144461_kernel(
    const float* __restrict__ x1, const float* __restrict__ x2,
    float* __restrict__ out) {
  __shared__ __align__(16) float A_lds[NPIXR * AS];        //  8.5 KB
  __shared__ __align__(16) float B_lds[NQR * BS];          // 34.8 KB
  __shared__ __align__(16) float D_lds[4 * NPIX * DSTR];   // 37.1 KB

  const int tid  = threadIdx.x;
  const int lane = tid & 31;
  const int wave = tid >> 5;
  const int wy   = wave >> 1;             // wave's 4x4 tile within 8x8 region
  const int wx   = wave & 1;
  const int py0  = wy * 4;
  const int px0  = wx * 4;

  const int TX = WDIM / REG;              // 16
  const int TY = HDIM / REG;              // 16
  int wg  = blockIdx.x;
  int b   = wg / (TX * TY);
  int rem = wg - b * (TX * TY);
  const int y0 = (rem / TX) * REG;
  const int x0 = (rem - (rem / TX) * TX) * REG;

  v8f acc[9] = {};                        // 9 N-tiles of 16 -> covers 144 q's

  // fragment geometry (f32 WMMA 16x16x4 layouts)
  const int pRow = lane & 15;             // M (A) / N (B)
  const int cSel = (lane >> 4) * 2;       // K sub-select
  const int pp   = (py0 + (pRow >> 2)) * REG + (px0 + (pRow & 3));
  const float* aBase = &A_lds[pp * AS + cSel];
  const float* bBase[9];
#pragma unroll
  for (int t = 0; t < 9; ++t) {
    int q  = t * 16 + pRow;               // wave-local patch index
    int qy = q / QW;
    int qx = q - qy * QW;
    int qq = (qy + py0) * QWR + (qx + px0);  // shared-patch index
    bBase[t] = &B_lds[qq * BS + cSel];
  }

  const size_t cstride = (size_t)HDIM * WDIM;   // channel stride (elements)

  for (int kc = 0; kc < NCHUNK; ++kc) {
    const int cbase = kc * KC;
    const bool pf = (kc + 1 < NCHUNK);

    // ---- stage A: x1 region [64 pix x KC ch] ----
#pragma unroll
    for (int it = 0; it < (NPIXR * KC) / 128; ++it) {   // 16 iters
      int idx = tid + it * 128;
      int p = idx & 63;
      int c = idx >> 6;
      int gy = y0 + (p >> 3);
      int gx = x0 + (p & 7);
      const float* src =
          &x1[(((size_t)b * CH + (cbase + c)) * HDIM + gy) * WDIM + gx];
      A_lds[p * AS + c] = *src;
      if (pf && (tid & 15) == 0) __builtin_prefetch(src + KC * cstride, 0, 3);
    }

    // ---- stage B: zero-padded shared x2 patch [256 q x KC ch] ----
#pragma unroll 4
    for (int it = 0; it < (NQR * KC) / 128; ++it) {     // 64 iters
      int idx = tid + it * 128;
      int q = idx & 255;
      int c = idx >> 8;
      int qy = q >> 4;
      int qx = q & 15;
      int gy = y0 + qy - MAXD;
      int gx = x0 + qx - MAXD;
      float v = 0.0f;
      if ((unsigned)gy < (unsigned)HDIM && (unsigned)gx < (unsigned)WDIM) {
        const float* src =
            &x2[(((size_t)b * CH + (cbase + c)) * HDIM + gy) * WDIM + gx];
        v = *src;
        if (pf && (tid & 15) == 0) __builtin_prefetch(src + KC * cstride, 0, 3);
      }
      B_lds[q * BS + c] = v;
    }
    __syncthreads();

    // ---- compute: 8 k-steps (K=4) x 9 q-tiles of fp32 WMMA ----
#pragma unroll
    for (int ks = 0; ks < KC / 4; ++ks) {
      v2f a = *(const v2f*)(aBase + ks * 4);       // A[M=pRow, K pair]
#pragma unroll
      for (int t = 0; t < 9; ++t) {
        v2f bb = *(const v2f*)(bBase[t] + ks * 4); // B[K pair, N=pRow]
        acc[t] = __builtin_amdgcn_wmma_f32_16x16x4_f32(
            /*neg_a=*/false, a, /*neg_b=*/false, bb,
            /*c_mod=*/(short)0, acc[t], /*reuse_a=*/false, /*reuse_b=*/false);
      }
    }
    __syncthreads();
  }

  // ---- spill D (scaled by 1/C) to this wave's LDS region ----
  {
    float* Dw = &D_lds[wave * NPIX * DSTR];
    const int mBase = (lane >> 4) * 8;
    const int n = lane & 15;
#pragma unroll
    for (int t = 0; t < 9; ++t) {
#pragma unroll
      for (int i = 0; i < 8; ++i) {
        Dw[(mBase + i) * DSTR + (t * 16 + n)] = acc[t][i] * (1.0f / CH);
      }
    }
  }
  __syncthreads();

  // ---- gather valid displacements, store 4-pixel bursts per plane ----
  {
    const float* Dw = &D_lds[wave * NPIX * DSTR];
    const int y0w = y0 + py0;
    const int x0w = x0 + px0;
    for (int o = lane; o < NOUT * NPIX; o += 32) {
      int d  = o >> 4;                    // displacement plane 0..80
      int p  = o & 15;                    // pixel in this wave's 4x4 tile
      int py = p >> 2, px = p & 3;
      int qy = py + d / NDISP;            // = py + (dy+4)
      int qx = px + d % NDISP;            // = px + (dx+4)
      float v = Dw[p * DSTR + qy * QW + qx];
      out[(((size_t)b * NOUT + d) * HDIM + (y0w + py)) * WDIM + (x0w + px)] = v;
    }
  }
}

extern "C" void kernel_launch(void* const* d_in, const int* in_sizes, int n_in,
                              void* d_out, int out_size, void* d_ws, size_t ws_size,
                              hipStream_t stream) {
  (void)in_sizes; (void)n_in; (void)d_ws; (void)ws_size; (void)out_size;
  const float* x1 = (const float*)d_in[0];
  const float* x2 = (const float*)d_in[1];
  float* out = (float*)d_out;
  const int nblocks = BATCH * (HDIM / REG) * (WDIM / REG);  // 1024
  ModuleCorrelation_34651796144461_kernel<<<nblocks, 128, 0, stream>>>(x1, x2, out);
}